// SSIM_53180285059357
// MI455X (gfx1250) — compile-verified
//
#include <hip/hip_runtime.h>
#include <math.h>

typedef __attribute__((ext_vector_type(2))) float v2f;
typedef __attribute__((ext_vector_type(8))) float v8f;

#define H 512
#define W 512
#define NIMG 48                         // 16 batch * 3 channels
#define NTILES (NIMG * 32 * 32)         // 49152 tiles of 16x16
#define INV_N (1.0f / 12582912.0f)      // 1 / (16*3*512*512)

__device__ __forceinline__ float loadz(const float* __restrict__ img, int r, int c) {
    // branchless zero-padded load (clamp address, mask value) -> no EXEC divergence
    int rc = min(max(r, 0), H - 1);
    int cc = min(max(c, 0), W - 1);
    float v = img[rc * W + cc];
    bool in = (r >= 0) & (r < H) & (c >= 0) & (c < W);
    return in ? v : 0.0f;
}

// Horizontal pass: Ih = Xpatch * Wh (banded Gaussian as B operand).
// GUARD=false: interior tile, raw loads with immediate offsets (no pad checks).
// GUARD=true : border tile, clamp+mask zero padding.
template <bool GUARD>
__device__ __forceinline__ void horiz_pass(const float* __restrict__ xi,
                                           const float* __restrict__ yi,
                                           const v2f wB[7],
                                           float ihT[5][16][28],
                                           int row0, int col0, int idx15, int half)
{
#pragma unroll
    for (int mb = 0; mb < 2; ++mb) {
        v8f hx = {}, hy = {}, hxx = {}, hyy = {}, hxy = {};
        const int grow = row0 - 5 + mb * 16 + idx15;          // patch row = A's M
        // base pointers for the interior fast path (immediate-offset loads)
        const float* __restrict__ px = xi + (size_t)((grow & 1023)) * W + col0 - 5 + 2 * half;
        const float* __restrict__ py = yi + (size_t)((grow & 1023)) * W + col0 - 5 + 2 * half;
#pragma unroll
        for (int c = 0; c < 7; ++c) {
            float ax0, ax1, ay0, ay1;
            if (GUARD) {
                const int gc0 = col0 - 5 + 4 * c + 2 * half;  // patch col = K
                ax0 = loadz(xi, grow, gc0);
                ax1 = loadz(xi, grow, gc0 + 1);
                ay0 = loadz(yi, grow, gc0);
                ay1 = loadz(yi, grow, gc0 + 1);
            } else {
                ax0 = px[4 * c];
                ax1 = px[4 * c + 1];
                ay0 = py[4 * c];
                ay1 = py[4 * c + 1];
            }
            v2f ax, ay, axx, ayy, axy;
            ax.x  = ax0;       ax.y  = ax1;
            ay.x  = ay0;       ay.y  = ay1;
            axx.x = ax0 * ax0; axx.y = ax1 * ax1;
            ayy.x = ay0 * ay0; ayy.y = ay1 * ay1;
            axy.x = ax0 * ay0; axy.y = ax1 * ay1;
            hx  = __builtin_amdgcn_wmma_f32_16x16x4_f32(false, ax,  false, wB[c], (short)0, hx,  false, false);
            hy  = __builtin_amdgcn_wmma_f32_16x16x4_f32(false, ay,  false, wB[c], (short)0, hy,  false, false);
            hxx = __builtin_amdgcn_wmma_f32_16x16x4_f32(false, axx, false, wB[c], (short)0, hxx, false, false);
            hyy = __builtin_amdgcn_wmma_f32_16x16x4_f32(false, ayy, false, wB[c], (short)0, hyy, false, false);
            hxy = __builtin_amdgcn_wmma_f32_16x16x4_f32(false, axy, false, wB[c], (short)0, hxy, false, false);
        }
        // C layout: VGPR v holds row (mb*16 + half*8 + v), col idx15.
        // Store transposed so the vertical pass can fetch B rows contiguously.
#pragma unroll
        for (int v = 0; v < 8; ++v) {
            int j = mb * 16 + half * 8 + v;
            if (j < 28) {
                ihT[0][idx15][j] = hx[v];
                ihT[1][idx15][j] = hy[v];
                ihT[2][idx15][j] = hxx[v];
                ihT[3][idx15][j] = hyy[v];
                ihT[4][idx15][j] = hxy[v];
            }
        }
    }
}

// One wave (32 threads) per 16x16 output tile. Separable 11-tap Gaussian conv
// done as banded matmuls on the WMMA pipe: 105 v_wmma_f32_16x16x4_f32 per tile.
__global__ __launch_bounds__(32) void ssim_tile(const float* __restrict__ x,
                                                const float* __restrict__ y,
                                                const float* __restrict__ win,
                                                float* __restrict__ partials,
                                                float* __restrict__ out,
                                                int atomicMode)
{
    __shared__ float gS[16];                       // 1D gaussian taps
    __shared__ __align__(16) float ihT[5][16][28]; // horizontal results, transposed

    const int lane  = threadIdx.x;     // 0..31
    const int idx15 = lane & 15;       // M row (A) / N col (B,C)
    const int half  = lane >> 4;       // K half-select

    // Recover exact 1D taps from the provided 2D window: w2d = outer(g,g)
    if (lane < 16) {
        float g5 = sqrtf(win[5 * 11 + 5]);        // g[5]
        float v  = (lane < 11) ? (win[5 * 11 + lane] / g5) : 0.0f;
        gS[lane] = v;
    }
    __syncthreads();

    // Banded Gaussian operand. Identical per-lane values serve as the B matrix
    // (horizontal: Wh[j][c]=g[j-c]) and the A matrix (vertical: Wv[m][j]=g[j-m]),
    // since both index with (lane&15) and K = 4c + 2*half + e.
    v2f wB[7];
#pragma unroll
    for (int c = 0; c < 7; ++c) {
        int d0 = 4 * c + 2 * half - idx15;
        int d1 = d0 + 1;
        float w0 = ((unsigned)d0 <= 10u) ? gS[d0] : 0.0f;
        float w1 = ((unsigned)d1 <= 10u) ? gS[d1] : 0.0f;
        v2f w; w.x = w0; w.y = w1;
        wB[c] = w;
    }

    const int tile = blockIdx.x;
    const int img  = tile >> 10;          // 1024 tiles per channel-image
    const int t    = tile & 1023;
    const int ty   = t >> 5, tx = t & 31;
    const int row0 = ty * 16, col0 = tx * 16;
    const float* xi = x + (size_t)img * (H * W);
    const float* yi = y + (size_t)img * (H * W);

    // Workgroup-uniform interior test (SALU branch, EXEC stays all-ones so the
    // WMMA all-lanes requirement holds in both paths). Interior needs the full
    // 26(+junk row reads up to row0+26) x 28 patch in-bounds:
    //   rows: row0-5 >= 0 and row0+26 <= 511  -> ty in [1,30]
    //   cols: col0-5 >= 0 and col0+22 <= 511  -> tx in [1,30]
    const bool interior = (ty >= 1) & (ty <= 30) & (tx >= 1) & (tx <= 30);
    if (interior) horiz_pass<false>(xi, yi, wB, ihT, row0, col0, idx15, half);
    else          horiz_pass<true >(xi, yi, wB, ihT, row0, col0, idx15, half);

    __syncthreads();   // single-wave workgroup: lowers to waitcnt + s_nop barrier

    // ---- Vertical pass: O = Wv * Ih ----
    v8f ox = {}, oy = {}, oxx = {}, oyy = {}, oxy = {};
#pragma unroll
    for (int c = 0; c < 7; ++c) {
        const int j0 = 4 * c + 2 * half;                // 8B-aligned (even * 4B, pitch 112B)
        v2f bx  = *(const v2f*)&ihT[0][idx15][j0];
        v2f by  = *(const v2f*)&ihT[1][idx15][j0];
        v2f bxx = *(const v2f*)&ihT[2][idx15][j0];
        v2f byy = *(const v2f*)&ihT[3][idx15][j0];
        v2f bxy = *(const v2f*)&ihT[4][idx15][j0];
        ox  = __builtin_amdgcn_wmma_f32_16x16x4_f32(false, wB[c], false, bx,  (short)0, ox,  false, false);
        oy  = __builtin_amdgcn_wmma_f32_16x16x4_f32(false, wB[c], false, by,  (short)0, oy,  false, false);
        oxx = __builtin_amdgcn_wmma_f32_16x16x4_f32(false, wB[c], false, bxx, (short)0, oxx, false, false);
        oyy = __builtin_amdgcn_wmma_f32_16x16x4_f32(false, wB[c], false, byy, (short)0, oyy, false, false);
        oxy = __builtin_amdgcn_wmma_f32_16x16x4_f32(false, wB[c], false, bxy, (short)0, oxy, false, false);
    }

    // ---- SSIM map + wave reduction ----
    const float C1 = 1e-4f, C2 = 9e-4f;
    float s = 0.0f;
#pragma unroll
    for (int v = 0; v < 8; ++v) {
        float mu1 = ox[v], mu2 = oy[v];
        float m11 = mu1 * mu1, m22 = mu2 * mu2, m12 = mu1 * mu2;
        float s1  = oxx[v] - m11;
        float s2  = oyy[v] - m22;
        float s12 = oxy[v] - m12;
        float num = (2.0f * m12 + C1) * (2.0f * s12 + C2);
        float den = (m11 + m22 + C1) * (s1 + s2 + C2);
        s += num / den;
    }
#pragma unroll
    for (int off = 16; off > 0; off >>= 1)
        s += __shfl_xor(s, off, 32);

    if (lane == 0) {
        if (atomicMode) atomicAdd(out, s * INV_N);
        else            partials[blockIdx.x] = s;
    }
}

// Deterministic single-block final reduction: partial sums -> mean
__global__ __launch_bounds__(256) void reduce_partials(const float* __restrict__ p,
                                                       float* __restrict__ out, int n)
{
    __shared__ float sm[256];
    float s = 0.0f;
    for (int i = threadIdx.x; i < n; i += 256) s += p[i];
    sm[threadIdx.x] = s;
    __syncthreads();
    for (int k = 128; k > 0; k >>= 1) {
        if ((int)threadIdx.x < k) sm[threadIdx.x] += sm[threadIdx.x + k];
        __syncthreads();
    }
    if (threadIdx.x == 0) out[0] = sm[0] * INV_N;
}

extern "C" void kernel_launch(void* const* d_in, const int* in_sizes, int n_in,
                              void* d_out, int out_size, void* d_ws, size_t ws_size,
                              hipStream_t stream) {
    const float* x   = (const float*)d_in[0];
    const float* y   = (const float*)d_in[1];
    const float* win = (const float*)d_in[2];   // [3,1,11,11], channel 0 used
    float* out = (float*)d_out;

    if (ws_size >= (size_t)NTILES * sizeof(float)) {
        float* partials = (float*)d_ws;
        ssim_tile<<<NTILES, 32, 0, stream>>>(x, y, win, partials, out, 0);
        reduce_partials<<<1, 256, 0, stream>>>(partials, out, NTILES);
    } else {
        hipMemsetAsync(d_out, 0, sizeof(float), stream);
        ssim_tile<<<NTILES, 32, 0, stream>>>(x, y, win, nullptr, out, 1);
    }
}